// PositionFeatures_10531259810360
// MI455X (gfx1250) — compile-verified
//
#include <hip/hip_runtime.h>

// ---------------------------------------------------------------------------
// PositionFeatures for MI455X (gfx1250, wave32).
//
// out[n,s,u,r] = sum_k F[s,u,r,k] * sum_c W[s,k,c] * bilinear(core[n,s,c], pos_u)
//
// Bilinear sampling commutes with the channel projection, so we sample core
// first (gather, L2-served: core = 170MB < 192MB L2) and run two small GEMMs
// on V_WMMA_F32_16X16X4_F32 (full fp32 precision; problem is memory-bound at
// ~1GB traffic / 23.3 TB/s, so low-precision WMMA buys nothing).
//
// Workspace: sampled[n][s][u][k] fp32 = 256 MiB.
// ---------------------------------------------------------------------------

typedef float v2f __attribute__((ext_vector_type(2)));
typedef float v4f __attribute__((ext_vector_type(4)));
typedef float v8f __attribute__((ext_vector_type(8)));

#define NN 32
#define SS 4
#define CC 256
#define KK 256
#define HH 36
#define WW 36
#define UU 2048
#define RR 32
#define UT 16          // u-tile per workgroup in kernel 1
#define LDSW 17        // padded row stride (floats) to break LDS bank conflicts

// ============================================================================
// Kernel 1: gather + projection GEMM.
//   block = (u-tile, s, n), 256 threads = 8 waves.
//   Phase A: cs[c][j] = bilinear(core[n,s,c], pos[u0+j]) into LDS.
//   Phase B: sampled[k][j] = sum_c W[s,k,c] * cs[c][j] via WMMA f32 16x16x4.
//   Phase C: transpose through LDS, NT-store sampled[n][s][u][k] (k-contig).
// ============================================================================
__global__ __launch_bounds__(256)
void pf_proj_sample_kernel(const float* __restrict__ core,   // (N, S*C, H, W)
                           const float* __restrict__ projw,  // (S, K, C)
                           const float* __restrict__ pos,    // (U, 2)
                           float* __restrict__ sampled)      // ws: (N, S, U, K)
{
    __shared__ float lds[CC * LDSW];   // 17 KB: cs, then reused for sampled

    const int u0  = blockIdx.x * UT;
    const int s   = blockIdx.y;
    const int n   = blockIdx.z;
    const int tid = threadIdx.x;

    // ---- Phase A: bilinear gather (border-clamped, matches reference) ----
    {
        const int j    = tid & 15;    // u within tile
        const int crow = tid >> 4;    // starting channel
        const int u    = u0 + j;
        const float gx = pos[2 * u + 0];
        const float gy = pos[2 * u + 1];
        float x = fminf(fmaxf(((gx + 1.0f) * (float)WW - 1.0f) * 0.5f, 0.0f), (float)(WW - 1));
        float y = fminf(fmaxf(((gy + 1.0f) * (float)HH - 1.0f) * 0.5f, 0.0f), (float)(HH - 1));
        const float x0f = floorf(x), y0f = floorf(y);
        const float wx = x - x0f, wy = y - y0f;
        const int x0 = (int)x0f, y0 = (int)y0f;
        const int x1 = min(x0 + 1, WW - 1), y1 = min(y0 + 1, HH - 1);
        const float w00 = (1.0f - wy) * (1.0f - wx);
        const float w01 = (1.0f - wy) * wx;
        const float w10 = wy * (1.0f - wx);
        const float w11 = wy * wx;
        const int o00 = y0 * WW + x0, o01 = y0 * WW + x1;
        const int o10 = y1 * WW + x0, o11 = y1 * WW + x1;
        const float* base = core + (size_t)((n * SS + s) * CC) * (HH * WW);
        for (int c = crow; c < CC; c += 16) {
            const float* p = base + (size_t)c * (HH * WW);
            lds[c * LDSW + j] =
                w00 * p[o00] + w01 * p[o01] + w10 * p[o10] + w11 * p[o11];
        }
    }
    __syncthreads();

    // ---- Phase B: WMMA f32 16x16x4, D[k,u] += W[k,c] * cs[c,u] ----
    // A 16x4 layout: lanes 0-15 -> M=lane, vgpr0=K0 vgpr1=K1; lanes 16-31 -> K2,K3.
    // B 4x16 layout: lanes 0-15 -> N=lane, vgpr0=K0 vgpr1=K1; lanes 16-31 -> K2,K3.
    const int lane = tid & 31;
    const int wave = tid >> 5;
    const int l16  = lane & 15;
    const int half = lane >> 4;
    const int k0a  = wave * 16;        // first 16x16 K-tile
    const int k0b  = wave * 16 + 128;  // second K-tile (shares B)
    const float* wA = projw + (size_t)s * KK * CC;

    v8f acc0 = {};
    v8f acc1 = {};
    for (int c = 0; c < CC; c += 4) {
        const int cp = c + 2 * half;                       // K0/K1 or K2/K3 pair
        v2f a0 = *(const v2f*)(wA + (size_t)(k0a + l16) * CC + cp);
        v2f a1 = *(const v2f*)(wA + (size_t)(k0b + l16) * CC + cp);
        v2f b;
        b.x = lds[cp * LDSW + l16];
        b.y = lds[(cp + 1) * LDSW + l16];
        acc0 = __builtin_amdgcn_wmma_f32_16x16x4_f32(false, a0, false, b,
                                                     (short)0, acc0, false, false);
        acc1 = __builtin_amdgcn_wmma_f32_16x16x4_f32(false, a1, false, b,
                                                     (short)0, acc1, false, false);
    }
    __syncthreads();   // done reading cs

    // ---- Phase C: D tiles -> LDS (sampled[k][j]), then coalesced NT store ----
    // C/D layout: vgpr i, lanes 0-15 -> M=i, lanes 16-31 -> M=i+8; N=lane%16.
#pragma unroll
    for (int i = 0; i < 8; ++i) {
        lds[(k0a + half * 8 + i) * LDSW + l16] = acc0[i];
        lds[(k0b + half * 8 + i) * LDSW + l16] = acc1[i];
    }
    __syncthreads();
    {
        const int j  = tid >> 4;          // u within tile
        const int kq = (tid & 15) * 16;   // 16 k's per thread
        float* dst = sampled + (((size_t)(n * SS + s) * UU) + u0 + j) * KK + kq;
#pragma unroll
        for (int q = 0; q < 16; q += 4) {
            v4f v;
            v.x = lds[(kq + q + 0) * LDSW + j];
            v.y = lds[(kq + q + 1) * LDSW + j];
            v.z = lds[(kq + q + 2) * LDSW + j];
            v.w = lds[(kq + q + 3) * LDSW + j];
            __builtin_nontemporal_store(v, (v4f*)(dst + q));   // streaming 256MiB
        }
    }
}

// ============================================================================
// Kernel 2: per-(s,u) feature contraction.
//   out[r=32, n=32] = F[s,u] (32x256) * sampled[s,u] (256x32)
//   block = 8 waves = 2 u's x 4 tiles (r0,n0 in {0,16}); 64 WMMA steps each.
// ============================================================================
__global__ __launch_bounds__(256)
void pf_feature_gemm_kernel(const float* __restrict__ feat,     // (S, U, R, K)
                            const float* __restrict__ sampled,  // (N, S, U, K)
                            float* __restrict__ out)            // (N, S, U, R)
{
    const int s    = blockIdx.y;
    const int tid  = threadIdx.x;
    const int wave = tid >> 5;
    const int lane = tid & 31;
    const int l16  = lane & 15;
    const int half = lane >> 4;
    const int u    = blockIdx.x * 2 + (wave >> 2);
    const int r0   = ((wave >> 1) & 1) * 16;
    const int n0   = (wave & 1) * 16;

    // A row (r0+l16) of F[s,u]; B column (n0+l16) of sampled; both k-contig.
    const float* A = feat + ((size_t)(s * UU + u) * RR + r0 + l16) * KK;
    const float* B = sampled + (((size_t)(n0 + l16) * SS + s) * UU + u) * KK;

    v8f acc = {};
    for (int c = 0; c < KK; c += 4) {
        const int cp = c + 2 * half;
        v2f a = *(const v2f*)(A + cp);
        v2f b = *(const v2f*)(B + cp);
        acc = __builtin_amdgcn_wmma_f32_16x16x4_f32(false, a, false, b,
                                                    (short)0, acc, false, false);
    }

    // D: lane -> column n=n0+l16, rows r0 + half*8 + i (i=0..7) -> r-contiguous.
    float* dst = out + (((size_t)(n0 + l16) * SS + s) * UU + u) * RR + r0 + half * 8;
    v4f lo, hi;
    lo.x = acc[0]; lo.y = acc[1]; lo.z = acc[2]; lo.w = acc[3];
    hi.x = acc[4]; hi.y = acc[5]; hi.z = acc[6]; hi.w = acc[7];
    __builtin_nontemporal_store(lo, (v4f*)dst);
    __builtin_nontemporal_store(hi, (v4f*)(dst + 4));
}

// ============================================================================
extern "C" void kernel_launch(void* const* d_in, const int* in_sizes, int n_in,
                              void* d_out, int out_size, void* d_ws, size_t ws_size,
                              hipStream_t stream) {
    const float* core      = (const float*)d_in[0];  // (N, S*C, H, W)
    const float* projw     = (const float*)d_in[1];  // (S, K, C)
    const float* features  = (const float*)d_in[2];  // (S, U, R, K)
    const float* positions = (const float*)d_in[3];  // (U, 2)
    float*       out       = (float*)d_out;          // (N, S, U, R)
    float*       sampled   = (float*)d_ws;           // (N, S, U, K) = 256 MiB

    dim3 g1(UU / UT, SS, NN);        // 128 x 4 x 32 workgroups
    pf_proj_sample_kernel<<<g1, 256, 0, stream>>>(core, projw, positions, sampled);

    dim3 g2(UU / 2, SS);             // 1024 x 4 workgroups
    pf_feature_gemm_kernel<<<g2, 256, 0, stream>>>(features, sampled, out);
}